// MOLAttn_72670846648423
// MI455X (gfx1250) — compile-verified
//
#include <hip/hip_runtime.h>
#include <math.h>

typedef __attribute__((ext_vector_type(2))) float v2f;
typedef __attribute__((ext_vector_type(8))) float v8f;

#define B_     64
#define T_     2000
#define DENC   512
#define IDIM_  512
#define HID_   256
#define K_     10
#define NSPLIT 20
#define ROWS_PER_SPLIT (T_ / NSPLIT)   // 100

// workspace layout (in floats)
#define WS_H     0
#define WS_MEANS (WS_H + B_ * HID_)        // 16384
#define WS_INVS  (WS_MEANS + B_ * K_)      // 17024
#define WS_PART  (WS_INVS + B_ * K_)       // 17664 (16B aligned)

// ---------------------------------------------------------------------------
// Kernel 1: h = tanh(state @ W1^T + b1) via V_WMMA_F32_16X16X4_F32.
// C tile grid: M = 64 (4 tiles), N = 256 (16 tiles) -> 64 waves total.
// A frag (16x4 f32): lane<16 -> K={kb,kb+1}; lane>=16 -> K={kb+2,kb+3}.
// B frag (4x16 f32): same K split; B[k][n] = W1[n][k] so it is a contiguous
// float2 along W1's row -> both frags are aligned b64 loads.
// ---------------------------------------------------------------------------
__global__ __launch_bounds__(256) void mlp1_wmma(const float* __restrict__ state,
                                                 const float* __restrict__ W1,
                                                 const float* __restrict__ b1,
                                                 float* __restrict__ h) {
  const int lane = threadIdx.x & 31;
  const int wave = threadIdx.x >> 5;
  const int tile = blockIdx.x * 8 + wave;     // 0..63
  const int m0   = (tile >> 4) << 4;          // 0,16,32,48
  const int n0   = (tile & 15) << 4;          // 0..240
  const int half = lane >> 4;                 // 0 or 1
  const int l16  = lane & 15;

  const float* arow = state + (size_t)(m0 + l16) * IDIM_ + 2 * half;
  const float* brow = W1    + (size_t)(n0 + l16) * IDIM_ + 2 * half;

  v8f c = {};
  #pragma unroll 4
  for (int kb = 0; kb < IDIM_; kb += 4) {
    v2f a = *(const v2f*)(arow + kb);
    v2f b = *(const v2f*)(brow + kb);
    c = __builtin_amdgcn_wmma_f32_16x16x4_f32(false, a, false, b,
                                              (short)0, c, false, false);
  }

  const float bias = b1[n0 + l16];
  #pragma unroll
  for (int j = 0; j < 8; ++j) {
    const int row = m0 + j + 8 * half;        // C/D VGPR layout
    h[(size_t)row * HID_ + n0 + l16] = tanhf(c[j] + bias);
  }
}

// ---------------------------------------------------------------------------
// Kernel 2: params = exp(h @ W2^T + b2); keep means (o%3==0) and 1/scales
// (o%3==1). weights (o%3==2) are softmaxed over a singleton axis in the
// reference -> identically 1.0, so they are dropped. One wave per batch row.
// ---------------------------------------------------------------------------
__global__ __launch_bounds__(32) void mlp2(const float* __restrict__ h,
                                           const float* __restrict__ W2,
                                           const float* __restrict__ b2,
                                           float* __restrict__ means,
                                           float* __restrict__ invs) {
  const int b = blockIdx.x;
  const int o = threadIdx.x;
  if (o >= 3 * K_) return;
  const float* hrow = h  + (size_t)b * HID_;
  const float* wrow = W2 + (size_t)o * HID_;
  float acc = b2[o];
  #pragma unroll 4
  for (int i = 0; i < HID_; ++i) acc += hrow[i] * wrow[i];
  const float p = expf(acc);
  const int k = o / 3;
  const int r = o - 3 * k;
  if (r == 0)      means[b * K_ + k] = p;
  else if (r == 1) invs [b * K_ + k] = 1.0f / p;
}

// ---------------------------------------------------------------------------
// Kernel 3: align[b,t] = sum_k sigmoid((t+.5-m)*is) - sigmoid((t-.5-m)*is)
// (weights == 1). Written straight into the align region of d_out.
// ---------------------------------------------------------------------------
__global__ __launch_bounds__(256) void align_kernel(const float* __restrict__ means,
                                                    const float* __restrict__ invs,
                                                    float* __restrict__ align_out) {
  const int idx = blockIdx.x * 256 + threadIdx.x;
  if (idx >= B_ * T_) return;
  const int b = idx / T_;
  const float t = (float)(idx - b * T_);
  float a = 0.f;
  #pragma unroll
  for (int k = 0; k < K_; ++k) {
    const float m  = means[b * K_ + k];
    const float is = invs [b * K_ + k];
    const float x1 = (t + 0.5f - m) * is;
    const float x2 = (t - 0.5f - m) * is;
    a += 1.0f / (1.0f + expf(-x1)) - 1.0f / (1.0f + expf(-x2));
  }
  align_out[idx] = a;
}

// ---------------------------------------------------------------------------
// Kernel 4: partial context. This is the bandwidth-critical kernel: streams
// all 262 MB of enc_z exactly once. 20 T-splits x 64 batches = 1280 blocks,
// 128 threads, one float4 column per thread (coalesced b128 loads), with a
// global_prefetch of the row 8 iterations ahead.
// ---------------------------------------------------------------------------
__global__ __launch_bounds__(128) void ctx_partial(const float* __restrict__ enc_z,
                                                   const float* __restrict__ align_in,
                                                   float* __restrict__ part) {
  const int s   = blockIdx.x;
  const int b   = blockIdx.y;
  const int tid = threadIdx.x;
  const int t0  = s * ROWS_PER_SPLIT;

  const float* base = enc_z + ((size_t)b * T_ + t0) * DENC + tid * 4;
  const float* arow = align_in + (size_t)b * T_ + t0;

  float4 acc = make_float4(0.f, 0.f, 0.f, 0.f);
  #pragma unroll 4
  for (int r = 0; r < ROWS_PER_SPLIT; ++r) {
    if (r + 8 < ROWS_PER_SPLIT)
      __builtin_prefetch(base + (size_t)(r + 8) * DENC, 0, 0);
    const float  a = arow[r];
    const float4 v = *(const float4*)(base + (size_t)r * DENC);
    acc.x += a * v.x; acc.y += a * v.y; acc.z += a * v.z; acc.w += a * v.w;
  }
  *(float4*)(part + (size_t)(b * NSPLIT + s) * DENC + tid * 4) = acc;
}

// ---------------------------------------------------------------------------
// Kernel 5: deterministic ordered reduction of the 20 partials per batch.
// ---------------------------------------------------------------------------
__global__ __launch_bounds__(128) void ctx_reduce(const float* __restrict__ part,
                                                  float* __restrict__ ctx_out) {
  const int b   = blockIdx.x;
  const int tid = threadIdx.x;
  float4 acc = make_float4(0.f, 0.f, 0.f, 0.f);
  for (int s = 0; s < NSPLIT; ++s) {
    const float4 v = *(const float4*)(part + (size_t)(b * NSPLIT + s) * DENC + tid * 4);
    acc.x += v.x; acc.y += v.y; acc.z += v.z; acc.w += v.w;
  }
  *(float4*)(ctx_out + (size_t)b * DENC + tid * 4) = acc;
}

extern "C" void kernel_launch(void* const* d_in, const int* in_sizes, int n_in,
                              void* d_out, int out_size, void* d_ws, size_t ws_size,
                              hipStream_t stream) {
  const float* state = (const float*)d_in[0];   // (B,1,IDIM)
  const float* enc_z = (const float*)d_in[1];   // (B,T,DENC)
  const float* W1    = (const float*)d_in[2];   // (HID,IDIM)
  const float* b1    = (const float*)d_in[3];   // (HID)
  const float* W2    = (const float*)d_in[4];   // (3K,HID)
  const float* b2    = (const float*)d_in[5];   // (3K)

  float* out       = (float*)d_out;
  float* ctx_out   = out;                 // B*DENC floats (context)
  float* align_out = out + B_ * DENC;     // B*T floats   (align)

  float* ws    = (float*)d_ws;
  float* h     = ws + WS_H;
  float* means = ws + WS_MEANS;
  float* invs  = ws + WS_INVS;
  float* part  = ws + WS_PART;

  mlp1_wmma   <<<8, 256, 0, stream>>>(state, W1, b1, h);
  mlp2        <<<B_, 32, 0, stream>>>(h, W2, b2, means, invs);
  align_kernel<<<(B_ * T_ + 255) / 256, 256, 0, stream>>>(means, invs, align_out);
  ctx_partial <<<dim3(NSPLIT, B_), 128, 0, stream>>>(enc_z, align_out, part);
  ctx_reduce  <<<B_, 128, 0, stream>>>(part, ctx_out);
}